// ClaheNormalizer_38173669327565
// MI455X (gfx1250) — compile-verified
//
#include <hip/hip_runtime.h>
#include <hip/hip_bf16.h>

// ---------------------------------------------------------------------------
// ClaheNormalizer for MI455X (gfx1250).
//   res = x - box9(x);  out = res / max(sqrt(box9(res^2)), 0.02)
// HBM-bound (336 MB min traffic ~14.4us @ 23.3TB/s). Design keeps LDS bytes
// and VALU lane-ops under that roofline:
//  - 32x32 tile + 8px halo staged via CDNA5 async global->LDS (ASYNCcnt)
//  - all LDS phases operate on float4 (b128 ds ops), horizontal 9-sums use
//    in-register sliding (14 adds / 4 outputs)
//  - b128 non-temporal stores (write stream must not evict input from L2;
//    halo re-reads are L2 hits since the whole input fits in 192MB L2)
// ---------------------------------------------------------------------------

#define AS_GLB __attribute__((address_space(1)))
#define AS_LDS __attribute__((address_space(3)))

#if __has_builtin(__builtin_amdgcn_global_load_async_to_lds_b128) && \
    __has_builtin(__builtin_amdgcn_global_load_async_to_lds_b32)  && \
    __has_builtin(__builtin_amdgcn_s_wait_asynccnt)
#define USE_ASYNC_LDS 1
#else
#define USE_ASYNC_LDS 0
#endif

namespace {

constexpr int   T         = 32;           // output tile edge
constexpr int   HLO       = 8;            // x halo (4 res-halo + 4 mean window)
constexpr int   XW        = T + 2 * HLO;  // 48
constexpr int   RW        = T + 8;        // 40
constexpr int   NTHREADS  = 256;          // 8 wave32
constexpr float MIN_STDEV = 0.02f;
constexpr float INV81     = 1.0f / 81.0f;

typedef int   v4i __attribute__((ext_vector_type(4)));
typedef float v4f __attribute__((ext_vector_type(4)));

// JAX/np 'reflect' (edge not repeated): -1 -> 1, H -> H-2. Overshoot <= 8 << H.
__device__ __forceinline__ int refl(int i, int n) {
  i = (i < 0) ? -i : i;
  return (i >= n) ? (2 * n - 2 - i) : i;
}

__device__ __forceinline__ AS_GLB v4i* glb_v4(const void* p) {
  return (AS_GLB v4i*)(unsigned long long)p;
}
__device__ __forceinline__ AS_GLB int* glb_i(const void* p) {
  return (AS_GLB int*)(unsigned long long)p;
}
__device__ __forceinline__ AS_LDS v4i* lds_v4(void* p) {
  return (AS_LDS v4i*)(unsigned int)(unsigned long long)p;
}
__device__ __forceinline__ AS_LDS int* lds_i(void* p) {
  return (AS_LDS int*)(unsigned int)(unsigned long long)p;
}

// 9-wide sliding sums over 12 consecutive values held in 3 float4s.
__device__ __forceinline__ v4f slide9(v4f a, v4f b, v4f c) {
  const float s0 = ((a.x + a.y) + (a.z + a.w)) + ((b.x + b.y) + (b.z + b.w)) + c.x;
  const float s1 = s0 - a.x + c.y;
  const float s2 = s1 - a.y + c.z;
  const float s3 = s2 - a.z + c.w;
  return (v4f){s0, s1, s2, s3};
}

}  // namespace

__global__ __launch_bounds__(NTHREADS)
void clahe_norm_tile_kernel(const float* __restrict__ in,
                            float* __restrict__ out,
                            int H, int W) {
  __shared__ float xs[XW * XW];  // padded input tile           (9.2 KB)
  __shared__ float h1[XW * RW];  // horizontal 9-sum of x       (7.7 KB)
  __shared__ float rs[RW * RW];  // res = x - mean1 (4-halo)    (6.4 KB)
  __shared__ float h2[RW * T];   // horizontal 9-sum of res^2   (5.1 KB)

  const int    tid     = threadIdx.x;
  const int    tx0     = blockIdx.x * T;
  const int    ty0     = blockIdx.y * T;
  const size_t img_off = (size_t)blockIdx.z * (size_t)H * (size_t)W;
  const float* __restrict__ src = in + img_off;

  // ---- Phase 1: async-gather 48x48 halo tile into LDS (reflect at borders).
  constexpr int G = XW / 4;  // 12 groups of 4 floats per row
  for (int g = tid; g < XW * G; g += NTHREADS) {
    const int r   = g / G;
    const int c4  = (g - r * G) * 4;
    const int iy  = refl(ty0 + r - HLO, H);
    const int ixb = tx0 + c4 - HLO;
    float* dst = &xs[r * XW + c4];
    const float* srow = src + (size_t)iy * W;
    if (ixb >= 0 && ixb + 3 < W) {
#if USE_ASYNC_LDS
      __builtin_amdgcn_global_load_async_to_lds_b128(
          glb_v4(srow + ixb), lds_v4(dst), 0, 0);
#else
      *(v4f*)dst = *(const v4f*)(srow + ixb);
#endif
    } else {
#pragma unroll
      for (int j = 0; j < 4; ++j) {
        const int ix = refl(ixb + j, W);
#if USE_ASYNC_LDS
        __builtin_amdgcn_global_load_async_to_lds_b32(
            glb_i(srow + ix), lds_i(dst + j), 0, 0);
#else
        dst[j] = srow[ix];
#endif
      }
    }
  }
#if USE_ASYNC_LDS
  __builtin_amdgcn_s_wait_asynccnt(0);  // my wave's async copies landed
#endif
  __syncthreads();                      // everyone's copies visible

  // ---- Phase 2: horizontal 9-sum of x -> h1[48][40]. 480 float4 units.
  constexpr int G2 = RW / 4;  // 10
  for (int u = tid; u < XW * G2; u += NTHREADS) {
    const int r  = u / G2;
    const int c4 = (u - r * G2) * 4;
    const float* base = &xs[r * XW + c4];
    const v4f a = *(const v4f*)(base);
    const v4f b = *(const v4f*)(base + 4);
    const v4f c = *(const v4f*)(base + 8);
    *(v4f*)&h1[r * RW + c4] = slide9(a, b, c);
  }
  __syncthreads();

  // ---- Phase 3: vertical 9-sum -> mean1; res = x - mean1 (4-halo). 400 units.
  for (int u = tid; u < RW * G2; u += NTHREADS) {
    const int r  = u / G2;
    const int c4 = (u - r * G2) * 4;
    v4f s = *(const v4f*)&h1[r * RW + c4];
#pragma unroll
    for (int k = 1; k < 9; ++k) s += *(const v4f*)&h1[(r + k) * RW + c4];
    const v4f ctr = *(const v4f*)&xs[(r + 4) * XW + (c4 + 4)];
    *(v4f*)&rs[r * RW + c4] = ctr - s * INV81;
  }
  __syncthreads();

  // ---- Phase 4: horizontal 9-sum of res^2 -> h2[40][32]. 320 units.
  constexpr int G4 = T / 4;  // 8
  for (int u = tid; u < RW * G4; u += NTHREADS) {
    const int r  = u / G4;
    const int c4 = (u - r * G4) * 4;
    const float* base = &rs[r * RW + c4];
    v4f a = *(const v4f*)(base);
    v4f b = *(const v4f*)(base + 4);
    v4f c = *(const v4f*)(base + 8);
    a *= a; b *= b; c *= c;
    *(v4f*)&h2[r * T + c4] = slide9(a, b, c);
  }
  __syncthreads();

  // ---- Phase 5: vertical 9-sum -> stdev; normalize; b128 NT store. 256 units.
  for (int u = tid; u < T * G4; u += NTHREADS) {
    const int r  = u / G4;
    const int c4 = (u - r * G4) * 4;
    v4f s = *(const v4f*)&h2[r * T + c4];
#pragma unroll
    for (int k = 1; k < 9; ++k) s += *(const v4f*)&h2[(r + k) * T + c4];
    const v4f ctr = *(const v4f*)&rs[(r + 4) * RW + (c4 + 4)];
    v4f o;
#pragma unroll
    for (int j = 0; j < 4; ++j) {
      const float sd = __builtin_sqrtf(s[j] * INV81);
      o[j] = ctr[j] / fmaxf(sd, MIN_STDEV);
    }
    // H, W are multiples of T (512/32), tiles are exact; 16B-aligned store.
    __builtin_nontemporal_store(
        o, (v4f*)(out + img_off + (size_t)(ty0 + r) * W + (tx0 + c4)));
  }
}

extern "C" void kernel_launch(void* const* d_in, const int* in_sizes, int n_in,
                              void* d_out, int out_size, void* d_ws, size_t ws_size,
                              hipStream_t stream) {
  (void)n_in; (void)out_size; (void)d_ws; (void)ws_size;
  const float* in  = (const float*)d_in[0];
  float*       out = (float*)d_out;

  const int H = 512, W = 512;
  const int nimg = in_sizes[0] / (H * W);  // B*C = 160

  dim3 grid((W + T - 1) / T, (H + T - 1) / T, nimg);
  dim3 block(NTHREADS);
  clahe_norm_tile_kernel<<<grid, block, 0, stream>>>(in, out, H, W);
}